// GenericGNNLayer_76381698392933
// MI455X (gfx1250) — compile-verified
//
#include <hip/hip_runtime.h>

typedef __attribute__((ext_vector_type(2))) float v2f;
typedef __attribute__((ext_vector_type(8))) float v8f;

#define N_NODES 50000
#define N_EDGES 800000
#define CH 96

// ---------------- degree histogram: degs[src[e]] += 1 ----------------
__global__ void degree_kernel(const int* __restrict__ src, float* __restrict__ degs) {
    int e = blockIdx.x * blockDim.x + threadIdx.x;
    if (e < N_EDGES) atomicAdd(&degs[src[e]], 1.0f);
}

// ---------------- norm[i] = clip(degs,1)^-0.5 ----------------
__global__ void norm_kernel(const float* __restrict__ degs, float* __restrict__ norm) {
    int i = blockIdx.x * blockDim.x + threadIdx.x;
    if (i < N_NODES) {
        float d = degs[i];
        norm[i] = rsqrtf(d < 1.0f ? 1.0f : d);
    }
}

// ---------------- edge aggregation: agg[dst] += feature[src] * norm[src] ----------------
// One wave32 per edge; each lane handles 3 channels (96 = 3*32). Gathers and
// atomic RMWs are per-instruction coalesced; feature/agg (19.2 MB each) are
// L2-resident on the 192 MB L2.
__global__ void scatter_kernel(const float* __restrict__ feature,
                               const int* __restrict__ src,
                               const int* __restrict__ dst,
                               const float* __restrict__ norm,
                               float* __restrict__ agg) {
    int e = blockIdx.x * (blockDim.x >> 5) + (threadIdx.x >> 5);
    int lane = threadIdx.x & 31;
    if (e >= N_EDGES) return;
    int s = src[e];
    int d = dst[e];
    float nrm = norm[s];
    const float* f = feature + (size_t)s * CH;
    float*       a = agg     + (size_t)d * CH;
    atomicAdd(&a[lane],       f[lane]       * nrm);
    atomicAdd(&a[lane + 32],  f[lane + 32]  * nrm);
    atomicAdd(&a[lane + 64],  f[lane + 64]  * nrm);
}

// ---------------- bias2[n] = b[n] + sum_k W[k][n]  (folds the "+1.0" through the GEMM) ----
__global__ void bias2_kernel(const float* __restrict__ W, const float* __restrict__ b,
                             float* __restrict__ bias2) {
    int n = threadIdx.x;
    if (n < CH) {
        float acc = 0.0f;
        for (int k = 0; k < CH; ++k) acc += W[k * CH + n];
        bias2[n] = b[n] + acc;
    }
}

// ---------------- out = agg @ W + bias2, fp32 WMMA (V_WMMA_F32_16X16X4_F32) ----------------
// One wave per 16-row block of the output: the 16x96 A tile is loaded ONCE into
// registers (24 x float2 per lane, exactly the ISA 16x4 fp32 A layout), then
// reused across all 6 N-tiles -> 6x less A traffic than tile-per-wave.
// 144 chained WMMAs per wave; W (36 KB) stays L0/L2-hot for every wave.
__global__ void gemm_wmma_kernel(const float* __restrict__ agg,
                                 const float* __restrict__ W,
                                 const float* __restrict__ bias2,
                                 float* __restrict__ out) {
    const int numMTiles = N_NODES / 16;  // 3125
    int wave = blockIdx.x * (blockDim.x >> 5) + (threadIdx.x >> 5);
    if (wave >= numMTiles) return;       // wave-uniform: EXEC stays all-1s

    int lane  = threadIdx.x & 31;
    int lidx  = lane & 15;            // A: row within tile; B/C: column within tile
    int khalf = (lane >> 4) << 1;     // lanes 0-15 -> K+{0,1}; lanes 16-31 -> K+{2,3}
    int m0 = wave * 16;

    // Load the full 16x96 A tile once: per-lane 24 contiguous float2 (b64) loads.
    // A 16x4 fp32 layout: lane = M, (v0,v1) = (K+khalf, K+khalf+1).
    v2f a[24];
    const float* arow = agg + (size_t)(m0 + lidx) * CH;
#pragma unroll
    for (int kk = 0; kk < 24; ++kk) {
        a[kk].x = arow[kk * 4 + khalf + 0];
        a[kk].y = arow[kk * 4 + khalf + 1];
    }

    int rbase = m0 + ((lane >> 4) << 3);  // C/D: VGPR i -> row i or i+8; col = lidx

#pragma unroll
    for (int nt = 0; nt < 6; ++nt) {
        int n0 = nt * 16;

        // Init accumulator with bias2 broadcast down each column.
        float bv = bias2[n0 + lidx];
        v8f c;
#pragma unroll
        for (int i = 0; i < 8; ++i) c[i] = bv;

#pragma unroll
        for (int kk = 0; kk < 24; ++kk) {
            v2f bm;
            // B 4x16 fp32 layout (K x N, row-major W[k*CH+n]): lane = N
            bm.x = W[(kk * 4 + khalf + 0) * CH + n0 + lidx];
            bm.y = W[(kk * 4 + khalf + 1) * CH + n0 + lidx];
            // 8 args: (neg_a, A, neg_b, B, c_mod, C, reuse_a, reuse_b)
            c = __builtin_amdgcn_wmma_f32_16x16x4_f32(false, a[kk], false, bm,
                                                      (short)0, c, false, false);
        }

#pragma unroll
        for (int i = 0; i < 8; ++i)
            out[(size_t)(rbase + i) * CH + n0 + lidx] = c[i];
    }
}

extern "C" void kernel_launch(void* const* d_in, const int* in_sizes, int n_in,
                              void* d_out, int out_size, void* d_ws, size_t ws_size,
                              hipStream_t stream) {
    (void)in_sizes; (void)n_in; (void)out_size; (void)ws_size;
    const float* feature = (const float*)d_in[0];
    const int*   src     = (const int*)d_in[1];
    const int*   dst     = (const int*)d_in[2];
    const float* W       = (const float*)d_in[3];
    const float* b       = (const float*)d_in[4];
    float*       out     = (float*)d_out;

    // Workspace layout (floats), 128-float aligned sections:
    float* ws    = (float*)d_ws;
    float* degs  = ws;                 // [0, 50000)        pad -> 50176
    float* norm  = ws + 50176;         // [50176, 100176)   pad -> 100352
    float* bias2 = ws + 100352;        // [100352, 100448)  pad -> 100480
    float* agg   = ws + 100480;        // [100480, 100480 + 4'800'000)

    size_t zero_bytes = (size_t)(100480 + (size_t)N_NODES * CH) * sizeof(float);
    hipMemsetAsync(d_ws, 0, zero_bytes, stream);  // zero degs + agg (capturable)

    degree_kernel<<<N_EDGES / 256, 256, 0, stream>>>(src, degs);
    norm_kernel<<<(N_NODES + 255) / 256, 256, 0, stream>>>(degs, norm);
    scatter_kernel<<<N_EDGES / 8, 256, 0, stream>>>(feature, src, dst, norm, agg);
    bias2_kernel<<<1, 128, 0, stream>>>(W, b, bias2);

    int totalWaves = N_NODES / 16;                      // 3125 row-block waves
    int blocks = (totalWaves + 7) / 8;                  // 8 waves / 256-thread block
    gemm_wmma_kernel<<<blocks, 256, 0, stream>>>(agg, W, bias2, out);
}